// UTopLayer_11295763988480
// MI455X (gfx1250) — compile-verified
//
#include <hip/hip_runtime.h>
#include <stdint.h>

// amdgpu-toolchain (clang-23) tensor_load_to_lds signature (6 args):
//   (u32x4 g0, i32x8 g1, i32x4 g2, i32x4 g3, i32x8 extra, i32 cpol)
typedef __attribute__((ext_vector_type(4))) unsigned int u32x4;
typedef __attribute__((ext_vector_type(8))) int          i32x8;
typedef __attribute__((ext_vector_type(4))) int          i32x4;

#define WAVE 32

// ---------------------------------------------------------------------------
// 1) Transpose inputs[64, N] -> xT[N, 64] using the Tensor Data Mover.
//    TDM pulls a 64x64 f32 tile (row stride N) into LDS, inserting 1 dword of
//    pad after every 64 dwords (row pitch 65) so the transposed LDS reads are
//    bank-conflict free. One TDM issue per workgroup (wave 0).
// ---------------------------------------------------------------------------
__global__ __launch_bounds__(256) void xT_kernel(const float* __restrict__ x,
                                                 float* __restrict__ xT,
                                                 int N) {
  __shared__ float tile[64 * 65];  // 16.6 KB, padded pitch written by TDM

  int j0 = blockIdx.x * 64;
  if (j0 > N - 64) j0 = N - 64;    // clamp tail tile (overlapping rewrite of
                                   // identical values is benign)

  if (threadIdx.x < WAVE) {        // wave 0 issues the DMA
    uint64_t gaddr = (uint64_t)(uintptr_t)x + (uint64_t)j0 * 4u;
    uint32_t lds   = (uint32_t)(uintptr_t)&tile[0];  // low 32 bits = LDS offset

    // ---- D# group 0 (128b): count | lds_addr | global_addr | type=2 ----
    u32x4 g0;
    g0[0] = 1u;                                        // count=1, not restore, no gather
    g0[1] = lds;                                       // lds_addr [63:32]
    g0[2] = (uint32_t)(gaddr & 0xFFFFFFFFull);         // global_addr[31:0]
    g0[3] = (uint32_t)((gaddr >> 32) & 0x01FFFFFFull)  // global_addr[56:32]
          | (2u << 30);                                // type = 2 ("image")

    // ---- D# group 1 (256b) ----
    const uint32_t data_size    = 2;  // 4-byte elements
    const uint32_t pad_enable   = 1;
    const uint32_t pad_interval = 5;  // pad after 64 dwords (2<<5)
    const uint32_t pad_amount   = 0;  // pad by 1 dword
    const uint32_t td0 = (uint32_t)N; // tensor_dim0 (elements)
    const uint32_t td1 = 64u;         // tensor_dim1 (rows)
    i32x8 g1;
    g1[0] = (int)((data_size << 16) | (pad_enable << 20) |
                  (pad_interval << 22) | (pad_amount << 25)); // wg_mask=0
    g1[1] = (int)((td0 & 0xFFFFu) << 16);                     // dim0 lo16 in [31:16]
    g1[2] = (int)(((td0 >> 16) & 0xFFFFu) | ((td1 & 0xFFFFu) << 16));
    g1[3] = (int)(((td1 >> 16) & 0xFFFFu) | (64u << 16));     // tile_dim0 = 64
    g1[4] = (int)(64u);                                       // tile_dim1=64, tile_dim2=0
    g1[5] = (int)(uint32_t)N;                                 // dim0_stride lo32
    g1[6] = 0;                                                // dim0_stride hi16, dim1_stride lo16
    g1[7] = 0;                                                // dim1_stride hi32

    i32x4 g2 = {0, 0, 0, 0};   // 2D tensor: groups 2/3 unused (dims 0)
    i32x4 g3 = {0, 0, 0, 0};
    i32x8 gx = {0, 0, 0, 0, 0, 0, 0, 0};  // extra group (clang-23 form): zeros

    __builtin_amdgcn_tensor_load_to_lds(g0, g1, g2, g3, gx, 0);
    __builtin_amdgcn_s_wait_tensorcnt((short)0);
  }
  __syncthreads();

  // tile[b*65 + jj] -> xT[(j0+jj)*64 + b] : coalesced 256B stores,
  // conflict-free LDS reads thanks to the 65-dword pitch.
  int b = threadIdx.x & 63;
  for (int jj = (int)(threadIdx.x >> 6); jj < 64; jj += 4) {
    xT[(size_t)(j0 + jj) * 64 + b] = tile[b * 65 + jj];
  }
}

// ---------------------------------------------------------------------------
// 2) CSR build: histogram -> exclusive scan -> bucket scatter
// ---------------------------------------------------------------------------
__global__ void zero_u32(uint32_t* __restrict__ p, int n) {
  int i = blockIdx.x * blockDim.x + threadIdx.x;
  if (i < n) p[i] = 0u;
}

__global__ void count_kernel(const int* __restrict__ I,
                             uint32_t* __restrict__ counts, int E) {
  int e = blockIdx.x * blockDim.x + threadIdx.x;
  if (e < E) atomicAdd(&counts[I[e]], 1u);
}

__global__ __launch_bounds__(1024) void scan_kernel(uint32_t* __restrict__ counts,
                                                    uint32_t* __restrict__ rowStart,
                                                    int N) {
  __shared__ uint32_t sh[1024];
  __shared__ uint32_t s_carry;
  const int tid = threadIdx.x;
  if (tid == 0) s_carry = 0u;
  __syncthreads();
  for (int base = 0; base < N; base += 1024) {
    const int idx = base + tid;
    const uint32_t v = (idx < N) ? counts[idx] : 0u;
    sh[tid] = v;
    __syncthreads();
    for (int off = 1; off < 1024; off <<= 1) {
      uint32_t t = (tid >= off) ? sh[tid - off] : 0u;
      __syncthreads();
      sh[tid] += t;
      __syncthreads();
    }
    const uint32_t incl  = sh[tid];
    const uint32_t carry = s_carry;
    if (idx < N) {
      rowStart[idx] = carry + incl - v;  // exclusive prefix
      counts[idx]   = 0u;                // reset for use as scatter cursor
    }
    __syncthreads();
    if (tid == 1023) s_carry = carry + incl;
    __syncthreads();
  }
  if (tid == 0) rowStart[N] = s_carry;
}

__global__ void scatter_kernel(const int* __restrict__ I, const int* __restrict__ J,
                               const float* __restrict__ W3,
                               const float* __restrict__ vel,
                               const uint32_t* __restrict__ rowStart,
                               uint32_t* __restrict__ cursor,
                               uint32_t* __restrict__ edgeJ,
                               float* __restrict__ edgeW, int E) {
  int e = blockIdx.x * blockDim.x + threadIdx.x;
  if (e >= E) return;
  const int i = I[e];
  const int j = J[e];
  const uint32_t pos = rowStart[i] + atomicAdd(&cursor[i], 1u);
  edgeJ[pos] = (uint32_t)j;
  edgeW[pos] = W3[e] * vel[j];  // fold velocity scale in once
}

// ---------------------------------------------------------------------------
// 3) One wave32 per output row: gather xT[j] (256B/wave), FMA, single store.
//    B = 64 -> 2 accumulators per lane. No float atomics anywhere.
// ---------------------------------------------------------------------------
__global__ __launch_bounds__(256) void spmm_kernel(const float* __restrict__ xT,
                                                   const uint32_t* __restrict__ rowStart,
                                                   const uint32_t* __restrict__ edgeJ,
                                                   const float* __restrict__ edgeW,
                                                   const float* __restrict__ bias,
                                                   float* __restrict__ out, int N) {
  const int row  = (int)((blockIdx.x * blockDim.x + threadIdx.x) >> 5);
  const int lane = (int)(threadIdx.x & 31u);
  if (row >= N) return;
  const uint32_t s  = rowStart[row];
  const uint32_t e1 = rowStart[row + 1];
  const float bi = bias[row];
  float a0 = bi, a1 = bi;
  for (uint32_t e = s; e < e1; ++e) {
    const uint32_t j = edgeJ[e];        // wave-uniform
    const float    w = edgeW[e];        // wave-uniform
    const float2 xv = ((const float2*)(xT + (size_t)j * 64))[lane];
    a0 = fmaf(xv.x, w, a0);
    a1 = fmaf(xv.y, w, a1);
  }
  out[(size_t)(2 * lane)     * N + row] = a0;
  out[(size_t)(2 * lane + 1) * N + row] = a1;
}

// ---------------------------------------------------------------------------
extern "C" void kernel_launch(void* const* d_in, const int* in_sizes, int n_in,
                              void* d_out, int out_size, void* d_ws, size_t ws_size,
                              hipStream_t stream) {
  const float* x    = (const float*)d_in[0];  // inputs [B, N]
  const float* W3   = (const float*)d_in[1];  // [E]
  const float* bia  = (const float*)d_in[2];  // [N]
  const float* vel  = (const float*)d_in[3];  // [N]
  const int*   I    = (const int*)d_in[4];    // [E]
  const int*   J    = (const int*)d_in[5];    // [E]
  float* out = (float*)d_out;

  const int N = in_sizes[2];   // bias length
  const int E = in_sizes[1];   // nnz

  // workspace carve-out (~19.6 MB): xT | rowStart | cursor | edgeJ | edgeW
  char* ws = (char*)d_ws;
  size_t off = 0;
  auto carve = [&](size_t bytes) -> void* {
    void* p = ws + off;
    off = (off + bytes + 255) & ~(size_t)255;
    return p;
  };
  float*    xT       = (float*)carve((size_t)N * 64 * sizeof(float));
  uint32_t* rowStart = (uint32_t*)carve((size_t)(N + 1) * sizeof(uint32_t));
  uint32_t* cursor   = (uint32_t*)carve((size_t)N * sizeof(uint32_t));
  uint32_t* edgeJ    = (uint32_t*)carve((size_t)E * sizeof(uint32_t));
  float*    edgeW    = (float*)carve((size_t)E * sizeof(float));
  (void)ws_size; (void)n_in; (void)out_size;

  const int nTiles = (N + 63) / 64;
  hipLaunchKernelGGL(xT_kernel, dim3(nTiles), dim3(256), 0, stream, x, xT, N);
  hipLaunchKernelGGL(zero_u32, dim3((N + 255) / 256), dim3(256), 0, stream, cursor, N);
  hipLaunchKernelGGL(count_kernel, dim3((E + 255) / 256), dim3(256), 0, stream, I, cursor, E);
  hipLaunchKernelGGL(scan_kernel, dim3(1), dim3(1024), 0, stream, cursor, rowStart, N);
  hipLaunchKernelGGL(scatter_kernel, dim3((E + 255) / 256), dim3(256), 0, stream,
                     I, J, W3, vel, rowStart, cursor, edgeJ, edgeW, E);
  hipLaunchKernelGGL(spmm_kernel, dim3((N + 7) / 8), dim3(256), 0, stream,
                     xT, rowStart, edgeJ, edgeW, bia, out, N);
}